// DAGNN_3066606649493
// MI455X (gfx1250) — compile-verified
//
#include <hip/hip_runtime.h>
#include <stdint.h>

// ---------------- problem constants (fixed by the reference) ----------------
#define N_NODES   100000
#define N_IN      1000
#define N_OUT     1000
#define N_EDGE    1600000
#define BATCH     32            // == wave32 lane count: lane = batch element
#define DEPTH     4
#define LVL_CHUNK 24750         // (N_NODES - N_IN) / DEPTH

// ---------------- scatter-kernel tiling ----------------
#define EDGE_CHUNK      128                       // edges per async refill: 3 x 512B
#define NCHUNK          (N_EDGE / EDGE_CHUNK)     // 12500 (exact)
#define WAVES_PER_BLOCK 8
#define SCATTER_BLOCKS  500
#define TOTAL_WAVES     (SCATTER_BLOCKS * WAVES_PER_BLOCK)  // 4000

// Issue one 512B async global->LDS copy: 32 lanes x 16B each.
__device__ __forceinline__ void async_copy_512(uint32_t lds_addr, uint64_t gaddr) {
    asm volatile("global_load_async_to_lds_b128 %0, %1, off"
                 :: "v"(lds_addr), "v"(gaddr) : "memory");
}

__device__ __forceinline__ uint32_t lds_addr_of(const void* p) {
    // low 32 bits of a generic pointer into LDS == wave-relative LDS byte address
    return (uint32_t)(uintptr_t)p;
}

// a_t layout: [node][lane]  (node-major so each edge touches one 128B line).
// Hot kernel first so the disasm snippet shows it.
__global__ __launch_bounds__(256) void k_scatter(const int*   __restrict__ W_src,
                                                 const int*   __restrict__ W_dst,
                                                 const float* __restrict__ W_val,
                                                 float*       __restrict__ a_t,
                                                 int lo /* first dst of this level */) {
    // per-wave double-buffered staging: [wave][buf][{src,dst,val}][128]
    __shared__ __align__(16) int stage[WAVES_PER_BLOCK][2][3][EDGE_CHUNK];

    const int lane    = threadIdx.x & 31;
    const int waveBlk = threadIdx.x >> 5;
    const int wave    = blockIdx.x * WAVES_PER_BLOCK + waveBlk;
    if (wave >= NCHUNK) return;

    const int laneByte = lane * 16;

    // kick off first refill into buffer 0
    {
        int c = wave;
        async_copy_512(lds_addr_of(&stage[waveBlk][0][0][0]) + laneByte,
                       (uint64_t)(W_src + c * EDGE_CHUNK) + laneByte);
        async_copy_512(lds_addr_of(&stage[waveBlk][0][1][0]) + laneByte,
                       (uint64_t)(W_dst + c * EDGE_CHUNK) + laneByte);
        async_copy_512(lds_addr_of(&stage[waveBlk][0][2][0]) + laneByte,
                       (uint64_t)(W_val + c * EDGE_CHUNK) + laneByte);
    }

    int cur = wave;
    int buf = 0;
    while (cur < NCHUNK) {
        int nxt = cur + TOTAL_WAVES;
        if (nxt < NCHUNK) {
            int nb = buf ^ 1;
            async_copy_512(lds_addr_of(&stage[waveBlk][nb][0][0]) + laneByte,
                           (uint64_t)(W_src + nxt * EDGE_CHUNK) + laneByte);
            async_copy_512(lds_addr_of(&stage[waveBlk][nb][1][0]) + laneByte,
                           (uint64_t)(W_dst + nxt * EDGE_CHUNK) + laneByte);
            async_copy_512(lds_addr_of(&stage[waveBlk][nb][2][0]) + laneByte,
                           (uint64_t)(W_val + nxt * EDGE_CHUNK) + laneByte);
            // 3 newer loads may stay outstanding; oldest 3 (current buffer) must land
            asm volatile("s_wait_asynccnt 0x3" ::: "memory");
        } else {
            asm volatile("s_wait_asynccnt 0x0" ::: "memory");
        }

        const int*   s_src = &stage[waveBlk][buf][0][0];
        const int*   s_dst = &stage[waveBlk][buf][1][0];
        const float* s_wv  = (const float*)&stage[waveBlk][buf][2][0];

        #pragma unroll 8
        for (int j = 0; j < EDGE_CHUNK; ++j) {
            int d = s_dst[j];                        // wave-uniform LDS broadcast
            if ((unsigned)(d - lo) < (unsigned)LVL_CHUNK) {   // edge belongs to this level
                int   s = s_src[j];
                float w = s_wv[j];
                float v = a_t[(size_t)s * BATCH + lane] * w;   // 128B coalesced gather
                atomicAdd(&a_t[(size_t)d * BATCH + lane], v);  // 128B coalesced atomic line
            }
        }
        cur = nxt;
        buf ^= 1;
    }
}

__global__ void k_init(const float* __restrict__ x, float* __restrict__ a_t) {
    int idx = blockIdx.x * blockDim.x + threadIdx.x;     // over N_NODES*32
    if (idx >= N_NODES * BATCH) return;
    int node = idx >> 5;
    int lane = idx & 31;
    // inputs clamped to x (transposed read, tiny); non-inputs start at 0 accumulator
    a_t[idx] = (node < N_IN) ? x[lane * N_IN + node] : 0.0f;
}

__global__ void k_activate(const float* __restrict__ bias,
                           float* __restrict__ a_t, int level) {
    int idx = blockIdx.x * blockDim.x + threadIdx.x;     // over LVL_CHUNK*32
    if (idx >= LVL_CHUNK * BATCH) return;
    int node = N_IN + level * LVL_CHUNK + (idx >> 5);
    int lane = idx & 31;
    size_t off = (size_t)node * BATCH + lane;
    float z = a_t[off] + bias[node];
    a_t[off] = 1.0f / (1.0f + __expf(-z));
}

__global__ void k_output(const float* __restrict__ a_t, float* __restrict__ out) {
    int idx = blockIdx.x * blockDim.x + threadIdx.x;     // over BATCH*N_OUT, [B,O] row-major
    if (idx >= BATCH * N_OUT) return;
    int b = idx / N_OUT;
    int o = idx % N_OUT;
    out[idx] = a_t[(size_t)(N_NODES - N_OUT + o) * BATCH + b];
}

extern "C" void kernel_launch(void* const* d_in, const int* in_sizes, int n_in,
                              void* d_out, int out_size, void* d_ws, size_t ws_size,
                              hipStream_t stream) {
    const float* x     = (const float*)d_in[0];   // [B, I]
    const int*   W_src = (const int*)  d_in[1];   // [E]
    const int*   W_dst = (const int*)  d_in[2];   // [E]
    const float* W_val = (const float*)d_in[3];   // [E]
    const float* bias  = (const float*)d_in[4];   // [N]

    float* a_t = (float*)d_ws;                    // [N][32] node-major, 12.8 MB

    k_init<<<(N_NODES * BATCH + 255) / 256, 256, 0, stream>>>(x, a_t);

    // exact DAG evaluation: one ordered pass per level (equivalent to 4 reference sweeps)
    for (int lvl = 0; lvl < DEPTH; ++lvl) {
        int lo = N_IN + lvl * LVL_CHUNK;
        k_scatter<<<SCATTER_BLOCKS, 256, 0, stream>>>(W_src, W_dst, W_val, a_t, lo);
        k_activate<<<(LVL_CHUNK * BATCH + 255) / 256, 256, 0, stream>>>(bias, a_t, lvl);
    }

    k_output<<<(BATCH * N_OUT + 255) / 256, 256, 0, stream>>>(a_t, (float*)d_out);
}